// CosSimSpatTempConvNet_41686952575611
// MI455X (gfx1250) — compile-verified
//
#include <hip/hip_runtime.h>

typedef __attribute__((ext_vector_type(2))) float v2f;
typedef __attribute__((ext_vector_type(4))) float v4f;
typedef __attribute__((ext_vector_type(8))) float v8f;

typedef __attribute__((address_space(3))) float lds_float;

#define NB     64      // batch
#define NCIN   64      // input channels
#define NT     4096    // time steps
#define NF     128     // filters
#define NK     64      // kernel width
#define NTOUT  4033    // NT - NK + 1
#define TT     64      // conv outputs per chunk
#define XW     128     // x chunk width (TT + NK)
#define XP     144     // padded x row stride in LDS (floats): half-lanes 32 banks apart
#define YP     132     // padded y row stride in LDS (floats): rows r, r+8 32 banks apart
#define NCHUNK 64      // ceil(NTOUT / TT)

// 32-bit LDS byte offset of a __shared__ element (generic -> local addrspacecast)
__device__ __forceinline__ unsigned lds_offset_of(const float* p) {
    return (unsigned)(uintptr_t)(lds_float*)p;
}

__global__ __launch_bounds__(256)
void cossim_fused_kernel(const float* __restrict__ x,     // [NB, NCIN, NT]
                         const float* __restrict__ cw,    // [NF, NK]
                         const float* __restrict__ sw,    // [NF, NCIN]
                         const float* __restrict__ wt,    // [NF]
                         float* __restrict__ part)        // [NB, NCHUNK]
{
    __shared__ float xs[NCIN * XP];    // 36864 B: x chunk, zero-filled past NT
    __shared__ float y_lds[NF * YP];   // 67584 B: y[f][j], j = t - t0
    __shared__ float sq[XW];           // per-t channel-sum of x^2
    __shared__ float sinv[TT];         // 1 / norm_inputs
    __shared__ float invnw[NF];        // 1 / norm_w
    __shared__ float red[256];

    const int chunk = blockIdx.x;
    const int b     = blockIdx.y;
    const int t0    = chunk * TT;
    const int tid   = threadIdx.x;

    const float* __restrict__ xb = x + (size_t)b * NCIN * NT;

    // ---- stage 0a: stage x chunk into LDS ----
    if (t0 + XW <= NT) {
        // uniform fast path: CDNA5 async global->LDS copy, 16 B per lane per op,
        // no VGPR round-trip; tracked with ASYNCcnt.
        for (int e = tid; e < NCIN * (XW / 4); e += 256) {
            const int c  = e >> 5;             // 32 vec4 per row
            const int j4 = (e & 31) << 2;
            const float* gaddr = xb + (size_t)c * NT + t0 + j4;
            const unsigned loff = lds_offset_of(&xs[c * XP + j4]);
            asm volatile("global_load_async_to_lds_b128 %0, %1, off"
                         :: "v"(loff), "v"(gaddr) : "memory");
        }
        asm volatile("s_wait_asynccnt 0x0" ::: "memory");
    } else {                           // tail chunk: per-element guard, zero-fill
        for (int e = tid; e < NCIN * XW; e += 256) {
            const int c = e >> 7, j = e & 127;
            const int t = t0 + j;
            xs[c * XP + j] = (t < NT) ? xb[(size_t)c * NT + t] : 0.0f;
        }
    }
    __syncthreads();

    // ---- stage 0b: sq[j] and 1/norm_w per filter ----
    if (tid < XW) {
        float s = 0.0f;
        for (int c = 0; c < NCIN; ++c) {
            const float v = xs[c * XP + tid];
            s += v * v;
        }
        sq[tid] = s;
        // tid doubles as filter index f (NF == 128)
        float s1 = 0.0f, s2 = 0.0f;
        for (int k = 0; k < NK; ++k)   { const float v = cw[tid * NK + k];   s1 += v * v; }
        for (int c = 0; c < NCIN; ++c) { const float v = sw[tid * NCIN + c]; s2 += v * v; }
        invnw[tid] = 1.0f / sqrtf(s1 * s2);   // ||merged_f|| = ||cw_f|| * ||sw_f||
    }
    __syncthreads();
    if (tid < TT) {
        float s = 0.0f;
        for (int k = 0; k < NK; ++k) s += sq[tid + k];
        sinv[tid] = (s > 0.0f) ? (1.0f / sqrtf(s)) : 0.0f;
    }
    __syncthreads();

    // ---- stage 1: y[f, t0+j] = sum_c sw[f,c] * xs[c, j] via WMMA f32 16x16x4 ----
    const int wave = tid >> 5;
    const int lane = tid & 31;
    const int half = lane >> 4;     // K-half select per ISA 16x4 f32 layout
    const int l16  = lane & 15;
    const int frow = wave * 16 + l16;

    // A fragments invariant across t-tiles: preload all 16 K-steps (32 VGPRs)
    v2f aval[16];
    #pragma unroll
    for (int st = 0; st < 16; ++st) {
        aval[st] = *(const v2f*)(sw + frow * NCIN + st * 4 + half * 2);
    }

    const float* __restrict__ bbase = &xs[(half * 2) * XP + l16];
    for (int tt = 0; tt < 8; ++tt) {
        v8f acc = {};
        const float* __restrict__ bp = bbase + tt * 16;
        #pragma unroll
        for (int st = 0; st < 16; ++st) {
            v2f bb;
            bb.x = bp[st * 4 * XP];          // channel st*4 + half*2
            bb.y = bp[st * 4 * XP + XP];     // channel st*4 + half*2 + 1
            acc = __builtin_amdgcn_wmma_f32_16x16x4_f32(
                false, aval[st], false, bb, (short)0, acc, false, false);
        }
        // D 16x16 f32: VGPR v holds rows v (lanes 0-15) and v+8 (lanes 16-31).
        // Single lane base + constant v*YP offsets -> ds_store immediate offsets.
        float* __restrict__ yb = &y_lds[(wave * 16 + half * 8) * YP + tt * 16 + l16];
        #pragma unroll
        for (int v = 0; v < 8; ++v) {
            yb[v * YP] = acc[v];
        }
    }
    __syncthreads();

    // ---- stage 2: temporal conv + cosine normalize + weighted |.| sum ----
    // Wave-uniform filter index: cw taps become SGPR (s_load) operands.
    // Each lane produces two adjacent outputs from one sliding v2f stream.
    float accv = 0.0f;
    const int i2 = lane << 1;          // even output index, 0..62
    for (int it = 0; it < 16; ++it) {
        const int f = __builtin_amdgcn_readfirstlane(wave + (it << 3));
        const float* __restrict__ cwg = cw + f * NK;   // scalar pointer -> s_load
        const float* __restrict__ yr  = &y_lds[f * YP + i2];
        float c0 = 0.0f, c1 = 0.0f;
        v2f p = *(const v2f*)yr;       // {y[i2], y[i2+1]}
        #pragma unroll
        for (int k = 0; k < NK; k += 2) {
            const v2f   n   = *(const v2f*)(yr + k + 2);
            const float ck  = cwg[k];
            const float ck1 = cwg[k + 1];
            c0 += ck * p.x + ck1 * p.y;        // output i2
            c1 += ck * p.y + ck1 * n.x;        // output i2+1
            p = n;
        }
        // scale = sqrt(NCIN*NK) = 64 exactly; invnw,sinv >= 0 so they factor
        // out of |.|; wt keeps its sign outside the abs.
        const float wfn = wt[f] * invnw[f] * 64.0f;
        const int t = t0 + i2;
        if (t < NTOUT)     accv += wfn * sinv[i2]     * fabsf(c0);
        if (t + 1 < NTOUT) accv += wfn * sinv[i2 + 1] * fabsf(c1);
    }
    red[tid] = accv;
    __syncthreads();
    for (int s = 128; s > 0; s >>= 1) {
        if (tid < s) red[tid] += red[tid + s];
        __syncthreads();
    }
    if (tid == 0) part[b * NCHUNK + chunk] = red[0];
}

// Deterministic final reduction: out[b] = sum(part)/Tout + sum(bias)
__global__ void cossim_final_kernel(const float* __restrict__ part,
                                    const float* __restrict__ bias,
                                    float* __restrict__ out)
{
    const int b = threadIdx.x;
    if (b < NB) {
        float s = 0.0f;
        for (int c = 0; c < NCHUNK; ++c) s += part[b * NCHUNK + c];
        float bs = 0.0f;
        for (int f = 0; f < NF; ++f) bs += bias[f];
        out[b] = s / (float)NTOUT + bs;
    }
}

extern "C" void kernel_launch(void* const* d_in, const int* in_sizes, int n_in,
                              void* d_out, int out_size, void* d_ws, size_t ws_size,
                              hipStream_t stream) {
    (void)in_sizes; (void)n_in; (void)out_size; (void)ws_size;
    const float* x    = (const float*)d_in[0];  // [64,64,4096]
    const float* cw   = (const float*)d_in[1];  // [128,64]
    const float* sw   = (const float*)d_in[2];  // [128,64]
    const float* wt   = (const float*)d_in[3];  // [128]
    const float* bias = (const float*)d_in[4];  // [128]
    float* out  = (float*)d_out;                // [64]
    float* part = (float*)d_ws;                 // [64,64] partials (16 KB)

    dim3 grid(NCHUNK, NB);
    cossim_fused_kernel<<<grid, 256, 0, stream>>>(x, cw, sw, wt, part);
    cossim_final_kernel<<<1, 64, 0, stream>>>(part, bias, out);
}